// LightGCN_34385508172283
// MI455X (gfx1250) — compile-verified
//
#include <hip/hip_runtime.h>
#include <hip/hip_bf16.h>
#include <math.h>

#define DIM      64
#define NLAYERS  3
#define DROP_P   0.1f
#define KEEP_SCALE 1.1111112f   // 1/(1-0.1)

typedef __attribute__((ext_vector_type(2))) float v2f;
typedef __attribute__((ext_vector_type(8))) float v8f;

// ---------------------------------------------------------------- dropout ---
__device__ __forceinline__ float hash_u01(unsigned int x) {
  x ^= 42u * 0x9E3779B9u;                 // fold in key(42)
  x = x * 747796405u + 2891336453u;       // PCG-ish mix
  x = ((x >> ((x >> 28) + 4u)) ^ x) * 277803737u;
  x = (x >> 22) ^ x;
  return (float)(x >> 8) * (1.0f / 16777216.0f);
}

__global__ __launch_bounds__(256) void k_dropout(const float* __restrict__ ev,
                                                 float* __restrict__ vals, int E) {
  int e = blockIdx.x * 256 + threadIdx.x;
  if (e >= E) return;
  vals[e] = (hash_u01((unsigned)e) >= DROP_P) ? ev[e] * KEEP_SCALE : 0.0f;
}

// ------------------------------------------------------------- gathers ------
__global__ __launch_bounds__(256) void k_gather0(const float* __restrict__ ue,
                                                 const float* __restrict__ ie,
                                                 const int* __restrict__ users,
                                                 const int* __restrict__ items,
                                                 float* __restrict__ ul,
                                                 float* __restrict__ il, int B) {
  int t = blockIdx.x * 256 + threadIdx.x;
  if (t >= B * DIM) return;
  int b = t >> 6, d = t & 63;
  ul[t] = ue[(size_t)users[b] * DIM + d];
  il[t] = ie[(size_t)items[b] * DIM + d];
}

__global__ __launch_bounds__(256) void k_gatherL(const float* __restrict__ emb,
                                                 const int* __restrict__ users,
                                                 const int* __restrict__ items,
                                                 float* __restrict__ ul,
                                                 float* __restrict__ il,
                                                 int B, int nUser) {
  int t = blockIdx.x * 256 + threadIdx.x;
  if (t >= B * DIM) return;
  int b = t >> 6, d = t & 63;
  ul[t] = emb[(size_t)users[b] * DIM + d];
  il[t] = emb[((size_t)nUser + (size_t)items[b]) * DIM + d];
}

// --------------------------------------------------------------- SpMM -------
// One wave32 per edge; each lane moves 2 consecutive floats (256B/edge).
__global__ __launch_bounds__(256) void k_spmm_first(const float* __restrict__ ue,
                                                    const float* __restrict__ ie,
                                                    const float* __restrict__ vals,
                                                    const int* __restrict__ src,
                                                    const int* __restrict__ dst,
                                                    float* __restrict__ out,
                                                    int E, int nUser) {
  unsigned t = blockIdx.x * 256u + threadIdx.x;
  int e = (int)(t >> 5);
  int lane = (int)(t & 31u);
  if (e >= E) return;
  float v = vals[e];
  if (v == 0.0f) return;                      // dropped edge
  int s = src[e];
  const float* srow = (s < nUser) ? (ue + (size_t)s * DIM)
                                  : (ie + (size_t)(s - nUser) * DIM);
  v2f x = *(const v2f*)(srow + lane * 2);
  float* drow = out + (size_t)dst[e] * DIM + lane * 2;
  atomicAdd(drow + 0, v * x.x);               // global_atomic_add_f32 (L2 RMW)
  atomicAdd(drow + 1, v * x.y);
}

__global__ __launch_bounds__(256) void k_spmm_next(const float* __restrict__ emb,
                                                   const float* __restrict__ vals,
                                                   const int* __restrict__ src,
                                                   const int* __restrict__ dst,
                                                   float* __restrict__ out, int E) {
  unsigned t = blockIdx.x * 256u + threadIdx.x;
  int e = (int)(t >> 5);
  int lane = (int)(t & 31u);
  if (e >= E) return;
  float v = vals[e];
  if (v == 0.0f) return;
  v2f x = *(const v2f*)(emb + (size_t)src[e] * DIM + lane * 2);
  float* drow = out + (size_t)dst[e] * DIM + lane * 2;
  atomicAdd(drow + 0, v * x.x);
  atomicAdd(drow + 1, v * x.y);
}

// ------------------------------------------------------------- epilogue -----
// light = mean over 4 layer embeddings; inter = light_u*light_i;
// inter_layers[b][l][d] = u_l * i_l
__global__ __launch_bounds__(256) void k_epilogue(const float* __restrict__ ul,
                                                  const float* __restrict__ il,
                                                  float* __restrict__ out_inter,
                                                  float* __restrict__ out_il, int B) {
  int t = blockIdx.x * 256 + threadIdx.x;
  if (t >= B * DIM) return;
  int b = t >> 6, d = t & 63;
  size_t stride = (size_t)B * DIM;
  float su = 0.f, si = 0.f;
#pragma unroll
  for (int l = 0; l <= NLAYERS; ++l) {
    float u = ul[l * stride + t];
    float i = il[l * stride + t];
    out_il[((size_t)b * (NLAYERS + 1) + l) * DIM + d] = u * i;
    su += u;
    si += i;
  }
  out_inter[t] = (su * 0.25f) * (si * 0.25f);
}

// ------------------------------------------------------- WMMA row-reduce ----
// ratings[b] = sigmoid(sum_d inter[b][d]) via V_WMMA_F32_16X16X4_F32 with an
// all-ones B matrix: D[m][n] = sum_k A[m][k], accumulated over 16 K=4 chunks.
// C layout: lane n (0-15) VGPR j = D[j][n]; lane n+16 VGPR j = D[j+8][n].
__global__ __launch_bounds__(256) void k_ratings_wmma(const float* __restrict__ inter,
                                                      float* __restrict__ ratings,
                                                      int B) {
  int wave = (blockIdx.x * 256 + threadIdx.x) >> 5;   // one 16-row tile / wave
  int lane = threadIdx.x & 31;
  int r0 = wave * 16;
  if (r0 >= B) return;

  if (r0 + 16 <= B) {                                  // wave-uniform branch
    int r = r0 + (lane & 15);                          // A: M = lane%16
    int colOff = (lane >> 4) * 2;                      // K = {0,1} or {2,3}
    const float* row = inter + (size_t)r * DIM + colOff;
    v8f c = {0.f, 0.f, 0.f, 0.f, 0.f, 0.f, 0.f, 0.f};
    v2f bones;
    bones.x = 1.0f;
    bones.y = 1.0f;
#pragma unroll
    for (int k = 0; k < 16; ++k) {
      v2f a = *(const v2f*)(row + 4 * k);              // A chunk 16x4
      c = __builtin_amdgcn_wmma_f32_16x16x4_f32(
          /*neg_a=*/false, a, /*neg_b=*/false, bones,
          /*c_mod=*/(short)0, c, /*reuse_a=*/false, /*reuse_b=*/false);
    }
    if ((lane & 15) == 0) {                            // column 0 holders
      int rb = r0 + (lane >> 4) * 8;
#pragma unroll
      for (int j = 0; j < 8; ++j)
        ratings[rb + j] = 1.0f / (1.0f + __expf(-c[j]));
    }
  } else {                                             // scalar tail tile
    if (lane < 16 && r0 + lane < B) {
      const float* row = inter + (size_t)(r0 + lane) * DIM;
      float s = 0.f;
      for (int d = 0; d < DIM; ++d) s += row[d];
      ratings[r0 + lane] = 1.0f / (1.0f + __expf(-s));
    }
  }
}

// ------------------------------------------------------------ launcher ------
extern "C" void kernel_launch(void* const* d_in, const int* in_sizes, int n_in,
                              void* d_out, int out_size, void* d_ws, size_t ws_size,
                              hipStream_t stream) {
  const float* uemb  = (const float*)d_in[0];
  const float* iemb  = (const float*)d_in[1];
  const float* evals = (const float*)d_in[2];
  const int*   esrc  = (const int*)d_in[3];
  const int*   edst  = (const int*)d_in[4];
  const int*   users = (const int*)d_in[5];
  const int*   items = (const int*)d_in[6];

  const int E     = in_sizes[2];
  const int B     = in_sizes[5];
  const int nUser = in_sizes[0] / DIM;
  const int nItem = in_sizes[1] / DIM;
  const size_t nNodes  = (size_t)nUser + (size_t)nItem;
  const size_t embByte = nNodes * DIM * sizeof(float);
  const size_t batByte = (size_t)(NLAYERS + 1) * B * DIM * sizeof(float);

  // workspace carve-out (256B aligned)
  char* w = (char*)d_ws;
  size_t off = 0;
  auto carve = [&](size_t bytes) {
    void* p = w + off;
    off = (off + bytes + 255) & ~(size_t)255;
    return p;
  };
  float* vals = (float*)carve((size_t)E * sizeof(float));
  float* ul   = (float*)carve(batByte);   // gathered user rows, 4 layers
  float* il   = (float*)carve(batByte);   // gathered item rows, 4 layers
  float* bufA = (float*)carve(embByte);   // ping
  float* bufB = (float*)carve(embByte);   // pong

  const int BD = B * DIM;
  const unsigned gE   = (unsigned)((E + 255) / 256);
  const unsigned gBD  = (unsigned)((BD + 255) / 256);
  const unsigned gSp  = (unsigned)(((size_t)E * 32 + 255) / 256);
  const unsigned gRt  = (unsigned)((((B + 15) / 16) + 7) / 8);

  // 1) edge dropout values
  k_dropout<<<gE, 256, 0, stream>>>(evals, vals, E);

  // 2) layer-0 gathered rows straight from inputs
  k_gather0<<<gBD, 256, 0, stream>>>(uemb, iemb, users, items, ul, il, B);

  // 3) three propagation layers, ping-pong full embedding buffers
  float* cur = nullptr;
  float* nxt = bufA;
  for (int l = 1; l <= NLAYERS; ++l) {
    hipMemsetAsync(nxt, 0, embByte, stream);
    if (l == 1)
      k_spmm_first<<<gSp, 256, 0, stream>>>(uemb, iemb, vals, esrc, edst, nxt, E, nUser);
    else
      k_spmm_next<<<gSp, 256, 0, stream>>>(cur, vals, esrc, edst, nxt, E);
    k_gatherL<<<gBD, 256, 0, stream>>>(nxt, users, items,
                                       ul + (size_t)l * BD, il + (size_t)l * BD,
                                       B, nUser);
    cur = nxt;
    nxt = (nxt == bufA) ? bufB : bufA;
  }

  // 4) outputs: ratings | inter | inter_layers
  float* out_ratings = (float*)d_out;
  float* out_inter   = out_ratings + B;
  float* out_il      = out_inter + (size_t)B * DIM;

  k_epilogue<<<gBD, 256, 0, stream>>>(ul, il, out_inter, out_il, B);
  k_ratings_wmma<<<gRt, 256, 0, stream>>>(out_inter, out_ratings, B);
}